// MVDR_18502719111751
// MI455X (gfx1250) — compile-verified
//
#include <hip/hip_runtime.h>
#include <math.h>

// ---------------- problem constants ----------------
#define WIN     4096
#define HOP     1024
#define NSAMP   32768
#define TFRAMES 29          // (32768-4096)/1024 + 1
#define FBINS   2049
#define NSIG    16          // BS(4) * C(4)

typedef __attribute__((ext_vector_type(2))) float v2f;
typedef __attribute__((ext_vector_type(8))) float v8f;

// ---------------- workspace layout (float offsets) ----------------
constexpr size_t OFF_WINDOW = 0;                         // 4096
constexpr size_t OFF_WSQ    = 4096;                      // 32768
constexpr size_t OFF_DFTR   = OFF_WSQ + 32768;           // 64x64
constexpr size_t OFF_DFTI   = OFF_DFTR + 4096;
constexpr size_t OFF_TWR    = OFF_DFTI + 4096;           // twiddle W_4096^{k1*n2}
constexpr size_t OFF_TWI    = OFF_TWR + 4096;
constexpr size_t SPECSZ     = (size_t)NSIG * TFRAMES * 4096;
constexpr size_t OFF_XR     = OFF_TWI + 4096;
constexpr size_t OFF_XI     = OFF_XR + SPECSZ;
constexpr size_t OFF_SR     = OFF_XI + SPECSZ;
constexpr size_t OFF_SI     = OFF_SR + SPECSZ;
constexpr size_t OFF_ER     = OFF_SI + SPECSZ;
constexpr size_t OFF_EI     = OFF_ER + SPECSZ;
constexpr size_t OFF_ACC    = OFF_EI + SPECSZ;           // 16*32768

// ------------------------------------------------------------------
// 16x16 output tile of a 64x64x64 f32 matmul on the WMMA pipe.
// A: row-major MxK matrix (lda), B: row-major KxN (ldb), bsign scales B.
// f32 A/B operand layout: lane L holds (A[M=m0+(L&15)][k+2h], A[..][k+2h+1]),
// h = L>>4; B mirrored. C/D: VGPR j -> M = j + 8h, N = lane&15.
// ------------------------------------------------------------------
__device__ __forceinline__ v8f wmma_acc(const float* A, int lda,
                                        const float* B, int ldb,
                                        int m0, int n0, float bsign, v8f acc) {
  const int lane = threadIdx.x & 31;
  const int r = lane & 15;
  const int h2 = (lane >> 4) << 1;   // 0 or 2
  for (int k = 0; k < 64; k += 4) {
    v2f a, b;
    a.x = A[(m0 + r) * lda + (k + h2)];
    a.y = A[(m0 + r) * lda + (k + h2 + 1)];
    b.x = bsign * B[(k + h2) * ldb + (n0 + r)];
    b.y = bsign * B[(k + h2 + 1) * ldb + (n0 + r)];
    acc = __builtin_amdgcn_wmma_f32_16x16x4_f32(false, a, false, b,
                                                (short)0, acc, false, false);
  }
  return acc;
}

// ---------------- K1: tables, wsq, zero OLA accumulator ----------------
__global__ void k_init(float* __restrict__ ws) {
  const int i = blockIdx.x * blockDim.x + threadIdx.x;  // up to 524288
  const float PI2 = 6.283185307179586f;
  if (i < 4096) {
    ws[OFF_WINDOW + i] = sqrtf(0.5f - 0.5f * cosf(PI2 * (float)i / 4096.0f));
    const int k = i >> 6, n = i & 63;
    const float a64 = PI2 * (float)((k * n) & 63) / 64.0f;
    ws[OFF_DFTR + i] = cosf(a64);          // W_64 = cos - i sin
    ws[OFF_DFTI + i] = -sinf(a64);
    const float a4k = PI2 * (float)((k * n) & 4095) / 4096.0f;
    ws[OFF_TWR + i] = cosf(a4k);           // W_4096^{k1*n2}
    ws[OFF_TWI + i] = -sinf(a4k);
  }
  if (i < NSAMP) {                         // overlap-add window-power denom
    float acc = 0.f;
    for (int t = 0; t < TFRAMES; ++t) {
      const int j = i - t * HOP;
      if (j >= 0 && j < WIN) {
        const float w = sqrtf(0.5f - 0.5f * cosf(PI2 * (float)j / 4096.0f));
        acc += w * w;
      }
    }
    ws[OFF_WSQ + i] = acc;
  }
  if (i < NSIG * NSAMP) ws[OFF_ACC + i] = 0.f;
}

// ---------------- K2: windowed STFT via two-stage 64x64 WMMA FFT ----------------
// n = 64*n1 + n2 ; k = k1 + 64*k2
// stage1: Y[k1][n2] = sum_n1 D[k1][n1] A[n1][n2]      (A real)
// twiddle: Y' = Y * W4096^{k1 n2}
// stage2: Z[k1][k2] = sum_n2 Y'[k1][n2] D[n2][k2]  -> X[k1 + 64 k2]
__global__ void __launch_bounds__(256)
k_stft(const float* __restrict__ x, const float* __restrict__ s,
       float* __restrict__ ws) {
  __shared__ float Af[4096];   // frame, then Y'_re
  __shared__ float Y2[4096];   // Y'_im
  const int sig = blockIdx.x / TFRAMES;   // 0..31
  const int t   = blockIdx.x % TFRAMES;

  const float* src;
  float *outR, *outI;
  if (sig < 16) {              // X side: x broadcast over S: x[(bs>>1)*4 + c]
    const int bs = sig >> 2, c = sig & 3;
    src  = x + (size_t)((bs >> 1) * 4 + c) * NSAMP;
    outR = ws + OFF_XR; outI = ws + OFF_XI;
  } else {                     // S side: flat (B,S,C,N) == sig-16 major
    src  = s + (size_t)(sig - 16) * NSAMP;
    outR = ws + OFF_SR; outI = ws + OFF_SI;
  }
  const size_t obase = ((size_t)(sig & 15) * TFRAMES + t) * 4096;

  const float* wdw = ws + OFF_WINDOW;
  for (int i = threadIdx.x; i < 4096; i += 256)
    Af[i] = src[t * HOP + i] * wdw[i];
  __syncthreads();

  const float* Dr = ws + OFF_DFTR;
  const float* Di = ws + OFF_DFTI;
  const float* Tr = ws + OFF_TWR;
  const float* Ti = ws + OFF_TWI;
  const int wave = threadIdx.x >> 5;
  const int lane = threadIdx.x & 31;
  const int r = lane & 15, hh = lane >> 4;

  // ---- stage 1: 16 tiles, 2 per wave ----
  v8f yr[2], yi[2];
  int m0s[2], n0s[2];
  for (int rep = 0; rep < 2; ++rep) {
    const int tid = wave * 2 + rep;
    const int m0 = (tid >> 2) * 16, n0 = (tid & 3) * 16;
    m0s[rep] = m0; n0s[rep] = n0;
    v8f a = {}; a = wmma_acc(Dr, 64, Af, 64, m0, n0, 1.f, a); yr[rep] = a;
    v8f b = {}; b = wmma_acc(Di, 64, Af, 64, m0, n0, 1.f, b); yi[rep] = b;
  }
  __syncthreads();
  // ---- twiddle, write back into LDS ----
  for (int rep = 0; rep < 2; ++rep)
    for (int j = 0; j < 8; ++j) {
      const int row = m0s[rep] + j + 8 * hh;
      const int col = n0s[rep] + r;
      const int idx = row * 64 + col;
      const float tr = Tr[idx], ti = Ti[idx];
      const float a = yr[rep][j], b = yi[rep][j];
      Af[idx] = a * tr - b * ti;
      Y2[idx] = a * ti + b * tr;
    }
  __syncthreads();
  // ---- stage 2 ----
  for (int rep = 0; rep < 2; ++rep) {
    const int tid = wave * 2 + rep;
    const int m0 = (tid >> 2) * 16, n0 = (tid & 3) * 16;
    v8f zr = {};
    zr = wmma_acc(Af, 64, Dr, 64, m0, n0,  1.f, zr);
    zr = wmma_acc(Y2, 64, Di, 64, m0, n0, -1.f, zr);
    v8f zi = {};
    zi = wmma_acc(Af, 64, Di, 64, m0, n0,  1.f, zi);
    zi = wmma_acc(Y2, 64, Dr, 64, m0, n0,  1.f, zi);
    for (int j = 0; j < 8; ++j) {
      const int k1 = m0 + j + 8 * hh;
      const int k2 = n0 + r;
      const size_t k = (size_t)k1 + 64 * (size_t)k2;
      outR[obase + k] = zr[j];
      outI[obase + k] = zi[j];
    }
  }
}

// ---------------- K3: causal PSD + Souden MVDR + filter apply ----------------
// one thread per (bs, f); sequential over T (causal cumsum)
__global__ void __launch_bounds__(256)
k_mvdr(float* __restrict__ ws) {
  const int idx = blockIdx.x * blockDim.x + threadIdx.x;
  if (idx >= 4 * FBINS) return;
  const int bs = idx / FBINS, f = idx % FBINS;

  const float* Xr = ws + OFF_XR; const float* Xi = ws + OFF_XI;
  const float* Sr = ws + OFF_SR; const float* Si = ws + OFF_SI;
  float* Er = ws + OFF_ER; float* Ei = ws + OFF_EI;

  float sSr[4][4] = {}, sSi[4][4] = {}, sNr[4][4] = {}, sNi[4][4] = {};

  for (int t = 0; t < TFRAMES; ++t) {
    float xr[4], xi[4], sr4[4], si4[4], nr[4], ni[4];
    for (int a = 0; a < 4; ++a) {
      const size_t ib = ((size_t)(bs * 4 + a) * TFRAMES + t) * 4096 + f;
      xr[a] = Xr[ib]; xi[a] = Xi[ib];
      sr4[a] = Sr[ib]; si4[a] = Si[ib];
      nr[a] = xr[a] - sr4[a]; ni[a] = xi[a] - si4[a];
    }
    // accumulate outer products  psd[a][b] = z_a * conj(z_b)
    for (int a = 0; a < 4; ++a)
      for (int b = 0; b < 4; ++b) {
        sSr[a][b] += sr4[a] * sr4[b] + si4[a] * si4[b];
        sSi[a][b] += si4[a] * sr4[b] - sr4[a] * si4[b];
        sNr[a][b] += nr[a] * nr[b] + ni[a] * ni[b];
        sNi[a][b] += ni[a] * nr[b] - nr[a] * ni[b];
      }
    const float icnt = 1.0f / (float)(t + 1);
    const float trn = (sNr[0][0] + sNr[1][1] + sNr[2][2] + sNr[3][3]) * icnt;
    const float eps = trn * 1e-7f + 1e-8f;

    // Gauss-Jordan inverse of (psd_n + eps I), 4x4 complex
    float Ar_[4][4], Ai_[4][4], Br_[4][4], Bi_[4][4];
    for (int a = 0; a < 4; ++a)
      for (int b = 0; b < 4; ++b) {
        Ar_[a][b] = sNr[a][b] * icnt + (a == b ? eps : 0.f);
        Ai_[a][b] = sNi[a][b] * icnt;
        Br_[a][b] = (a == b) ? 1.f : 0.f;
        Bi_[a][b] = 0.f;
      }
    for (int p = 0; p < 4; ++p) {
      const float pr = Ar_[p][p], pi = Ai_[p][p];
      const float d = 1.0f / (pr * pr + pi * pi + 1e-30f);
      const float ir = pr * d, ii = -pi * d;
      for (int c = 0; c < 4; ++c) {
        float a = Ar_[p][c], b = Ai_[p][c];
        Ar_[p][c] = a * ir - b * ii; Ai_[p][c] = a * ii + b * ir;
        a = Br_[p][c]; b = Bi_[p][c];
        Br_[p][c] = a * ir - b * ii; Bi_[p][c] = a * ii + b * ir;
      }
      for (int q = 0; q < 4; ++q) {
        if (q == p) continue;
        const float fr = Ar_[q][p], fi = Ai_[q][p];
        for (int c = 0; c < 4; ++c) {
          const float apr = Ar_[p][c], api = Ai_[p][c];
          const float bpr = Br_[p][c], bpi = Bi_[p][c];
          Ar_[q][c] -= fr * apr - fi * api;
          Ai_[q][c] -= fr * api + fi * apr;
          Br_[q][c] -= fr * bpr - fi * bpi;
          Bi_[q][c] -= fr * bpi + fi * bpr;
        }
      }
    }
    // num = inv(psd_n) @ psd_s ;  ws = num / (trace(num)+eps)
    float nmr[4][4], nmi[4][4];
    for (int a = 0; a < 4; ++a)
      for (int b = 0; b < 4; ++b) {
        float ar = 0.f, ai = 0.f;
        for (int k = 0; k < 4; ++k) {
          const float psr = sSr[k][b] * icnt, psi = sSi[k][b] * icnt;
          ar += Br_[a][k] * psr - Bi_[a][k] * psi;
          ai += Br_[a][k] * psi + Bi_[a][k] * psr;
        }
        nmr[a][b] = ar; nmi[a][b] = ai;
      }
    const float trr = nmr[0][0] + nmr[1][1] + nmr[2][2] + nmr[3][3] + 1e-8f;
    const float tri = nmi[0][0] + nmi[1][1] + nmi[2][2] + nmi[3][3];
    const float dd = 1.0f / (trr * trr + tri * tri);
    // enh[b] = sum_a conj(ws[a][b]) * X[a]
    for (int b = 0; b < 4; ++b) {
      float er = 0.f, ei = 0.f;
      for (int a = 0; a < 4; ++a) {
        const float wr = (nmr[a][b] * trr + nmi[a][b] * tri) * dd;
        const float wi = (nmi[a][b] * trr - nmr[a][b] * tri) * dd;
        er += wr * xr[a] + wi * xi[a];
        ei += wr * xi[a] - wi * xr[a];
      }
      const size_t ob = ((size_t)(bs * 4 + b) * TFRAMES + t) * 4096;
      Er[ob + f] = er; Ei[ob + f] = ei;
      if (f >= 1 && f <= 2047) {           // Hermitian extension for C2R iFFT
        Er[ob + 4096 - f] = er; Ei[ob + 4096 - f] = -ei;
      }
    }
  }
}

// ---------------- K4: iSTFT (conjugate two-stage WMMA FFT) + overlap-add ----------------
__global__ void __launch_bounds__(256)
k_istft(float* __restrict__ ws) {
  __shared__ float Ure[4096];   // spectrum matrix, then U'_re
  __shared__ float Uim[4096];
  const int sig = blockIdx.x / TFRAMES;   // 0..15
  const int t   = blockIdx.x % TFRAMES;
  const float* Er = ws + OFF_ER; const float* Ei = ws + OFF_EI;
  const size_t ibase = ((size_t)sig * TFRAMES + t) * 4096;

  // load the spectrum into LDS transposed: row k1, column k2 holds bin k1 + 64*k2
  for (int e = threadIdx.x; e < 4096; e += 256) {
    const int k1 = e >> 6, k2 = e & 63;
    Ure[e] = Er[ibase + k1 + 64 * k2];
    Uim[e] = Ei[ibase + k1 + 64 * k2];
  }
  __syncthreads();

  const float* Dr = ws + OFF_DFTR;
  const float* Di = ws + OFF_DFTI;
  const float* Tr = ws + OFF_TWR;
  const float* Ti = ws + OFF_TWI;
  const int wave = threadIdx.x >> 5;
  const int lane = threadIdx.x & 31;
  const int r = lane & 15, hh = lane >> 4;

  // stage A: U = Spec @ conj(D64)
  v8f ar[2], ai[2]; int m0s[2], n0s[2];
  for (int rep = 0; rep < 2; ++rep) {
    const int tid = wave * 2 + rep;
    const int m0 = (tid >> 2) * 16, n0 = (tid & 3) * 16;
    m0s[rep] = m0; n0s[rep] = n0;
    v8f ur = {};
    ur = wmma_acc(Ure, 64, Dr, 64, m0, n0,  1.f, ur);
    ur = wmma_acc(Uim, 64, Di, 64, m0, n0,  1.f, ur);
    v8f ui = {};
    ui = wmma_acc(Ure, 64, Di, 64, m0, n0, -1.f, ui);
    ui = wmma_acc(Uim, 64, Dr, 64, m0, n0,  1.f, ui);
    ar[rep] = ur; ai[rep] = ui;
  }
  __syncthreads();
  // conjugate twiddle, write back
  for (int rep = 0; rep < 2; ++rep)
    for (int j = 0; j < 8; ++j) {
      const int row = m0s[rep] + j + 8 * hh;
      const int col = n0s[rep] + r;
      const int idx = row * 64 + col;
      const float tr = Tr[idx], ti = Ti[idx];
      const float a = ar[rep][j], b = ai[rep][j];
      Ure[idx] = a * tr + b * ti;
      Uim[idx] = b * tr - a * ti;
    }
  __syncthreads();
  // stage B: frame = Re(conj(D64) @ U') / 4096, windowed overlap-add
  float* acc = ws + OFF_ACC;
  const float* wdw = ws + OFF_WINDOW;
  for (int rep = 0; rep < 2; ++rep) {
    const int tid = wave * 2 + rep;
    const int m0 = (tid >> 2) * 16, n0 = (tid & 3) * 16;
    v8f o = {};
    o = wmma_acc(Dr, 64, Ure, 64, m0, n0, 1.f, o);
    o = wmma_acc(Di, 64, Uim, 64, m0, n0, 1.f, o);
    for (int j = 0; j < 8; ++j) {
      const int n1 = m0 + j + 8 * hh;
      const int n2 = n0 + r;
      const int i = n1 * 64 + n2;
      const float v = o[j] * (1.0f / 4096.0f) * wdw[i];
      atomicAdd(acc + (size_t)sig * NSAMP + t * HOP + i, v);
    }
  }
}

// ---------------- K5: window-power normalization -> d_out ----------------
__global__ void k_final(const float* __restrict__ ws, float* __restrict__ out) {
  const int i = blockIdx.x * blockDim.x + threadIdx.x;
  if (i >= NSIG * NSAMP) return;
  const int n = i & (NSAMP - 1);
  out[i] = ws[OFF_ACC + i] / fmaxf(ws[OFF_WSQ + n], 1e-8f);
}

// ------------------------------------------------------------------
extern "C" void kernel_launch(void* const* d_in, const int* in_sizes, int n_in,
                              void* d_out, int out_size, void* d_ws, size_t ws_size,
                              hipStream_t stream) {
  const float* x = (const float*)d_in[0];   // (2,4,32768)
  const float* s = (const float*)d_in[1];   // (2,2,4,32768)
  float* out = (float*)d_out;               // (2,2,4,32768)
  float* ws  = (float*)d_ws;

  k_init <<<(NSIG * NSAMP + 255) / 256, 256, 0, stream>>>(ws);
  k_stft <<<32 * TFRAMES, 256, 0, stream>>>(x, s, ws);
  k_mvdr <<<(4 * FBINS + 255) / 256, 256, 0, stream>>>(ws);
  k_istft<<<NSIG * TFRAMES, 256, 0, stream>>>(ws);
  k_final<<<(NSIG * NSAMP + 255) / 256, 256, 0, stream>>>(ws, out);
}